// DCRNN_56453050139088
// MI455X (gfx1250) — compile-verified
//
#include <hip/hip_runtime.h>
#include <cmath>

typedef __attribute__((ext_vector_type(2))) float v2f;
typedef __attribute__((ext_vector_type(8))) float v8f;

#define B_SZ    64
#define N_NODES 1000
#define T_IN    12
#define T_OUT   12
#define HIDC    64
#define BN      (B_SZ * N_NODES)   /* 64000 rows; multiple of 64 */
#define CK      40                  /* K-chunk staged in LDS (1000 = 25*40) */
#define SP_LD   42                  /* padded LDS stride: even (8B align) + bank spread */

// =====================================================================
// Diffusion GEMM: y[b] = P @ x[b]   (P: N x N f32, L2-resident)
//   x: (B, N, ld_in) cols [0,C);  y: (B, N, ld_out) cols [0,C)
// grid = (ceil(N/64), 1, B)  block = (32, ceil(C/16))
// Block computes 64 x (16*blockDim.y) tile. P chunk (64 x CK) is staged
// in LDS once per block and shared by all waves; each wave runs 4
// accumulators over V_WMMA_F32_16X16X4_F32 with ds_load_b64 A-fragments.
// OOB rows/cols are ADDRESS-CLAMPED (never zero-selected): a junk A-row /
// B-col only corrupts the matching D-row/col, which is never stored.
// =====================================================================
__global__ __launch_bounds__(128)
void k_diffuse(const float* __restrict__ P,
               const float* __restrict__ xin, int ld_in,
               float* __restrict__ yout, int ld_out, int C) {
  __shared__ float sP[64 * SP_LD];

  const int lane = threadIdx.x;
  const int half = lane >> 4;        // 0: K pair {0,1}, 1: K pair {2,3}
  const int l16  = lane & 15;
  const int b    = blockIdx.z;
  const int row0 = blockIdx.x * 64;
  const int col  = threadIdx.y * 16 + l16;
  const int colc = (col < C) ? col : (C - 1);        // clamp junk cols
  const int tid  = threadIdx.y * 32 + threadIdx.x;
  const int nthr = blockDim.x * blockDim.y;

  const float* xb = xin  + (size_t)b * N_NODES * ld_in;
  float*       yb = yout + (size_t)b * N_NODES * ld_out;

  // LDS A-fragment bases: row (16s + l16), even dword offset -> ds_load_b64
  const float* sA[4];
#pragma unroll
  for (int s = 0; s < 4; ++s)
    sA[s] = sP + (16 * s + l16) * SP_LD + 2 * half;

  const float* pB = xb + (2 * half) * ld_in + colc;  // advances 4*ld_in / step

  v8f acc[4] = {};
  for (int kc = 0; kc < N_NODES; kc += CK) {
    __syncthreads();
    // cooperative stage of P[row0:row0+64, kc:kc+CK] (rows clamped)
    for (int idx = tid; idx < 64 * CK; idx += nthr) {
      int r = idx / CK, c = idx - r * CK;
      int gr = row0 + r;
      if (gr >= N_NODES) gr = N_NODES - 1;
      sP[r * SP_LD + c] = P[gr * N_NODES + kc + c];
    }
    __syncthreads();
#pragma unroll
    for (int k0 = 0; k0 < CK; k0 += 4) {
      v2f bf;
      bf.x = pB[0];
      bf.y = pB[ld_in];
      pB += 4 * ld_in;
#pragma unroll
      for (int s = 0; s < 4; ++s) {
        v2f a = *(const v2f*)(sA[s] + k0);           // ds_load_b64
        acc[s] = __builtin_amdgcn_wmma_f32_16x16x4_f32(
            false, a, false, bf, (short)0, acc[s], false, false);
      }
    }
  }

  if (col >= C) return;                              // after all WMMAs
  if (row0 + 64 <= N_NODES) {                        // fast path: full tile
#pragma unroll
    for (int s = 0; s < 4; ++s)
#pragma unroll
      for (int r = 0; r < 8; ++r)
        yb[(row0 + 16 * s + 8 * half + r) * ld_out + colc] = acc[s][r];
  } else {
#pragma unroll
    for (int s = 0; s < 4; ++s)
#pragma unroll
      for (int r = 0; r < 8; ++r) {
        int m = row0 + 16 * s + 8 * half + r;
        if (m < N_NODES) yb[m * ld_out + colc] = acc[s][r];
      }
  }
}

// =====================================================================
// Weight-apply GEMM with fused epilogue:
//   mode 0: out = A@W + bias
//   mode 1: out = sigmoid(out + A@W + bias)
//   mode 2: out = tanh   (out + A@W + bias)
// A: (BN, KD) contiguous; W: (KD, CO); out: (BN, CO). BN % 64 == 0.
// Main loop over KD&~3 is select-free; K-tail (KD=5 case) zero-selects.
// grid = (BN/64, 1, 1)  block = (32, ceil(CO/16))
// =====================================================================
__global__ __launch_bounds__(128)
void k_gemm_bias_act(const float* __restrict__ A, int KD,
                     const float* __restrict__ W,
                     const float* __restrict__ bias, int CO,
                     float* __restrict__ out, int mode) {
  const int lane = threadIdx.x;
  const int half = lane >> 4;
  const int l16  = lane & 15;
  const int row0 = blockIdx.x * 64;
  const int col  = threadIdx.y * 16 + l16;
  const int colc = (col < CO) ? col : (CO - 1);      // clamp junk cols

  const float* arow[4];
  const float* pA[4];
#pragma unroll
  for (int s = 0; s < 4; ++s) {
    arow[s] = A + (row0 + 16 * s + l16) * KD;
    pA[s]   = arow[s] + 2 * half;
  }
  const float* pW = W + (2 * half) * CO + colc;

  v8f acc[4] = {};
  const int KD4 = KD & ~3;
  for (int k0 = 0; k0 < KD4; k0 += 4) {
    v2f bf;
    bf.x = pW[0];
    bf.y = pW[CO];
    pW += 4 * CO;
#pragma unroll
    for (int s = 0; s < 4; ++s) {
      v2f a;
      a.x = pA[s][0];
      a.y = pA[s][1];
      pA[s] += 4;
      acc[s] = __builtin_amdgcn_wmma_f32_16x16x4_f32(
          false, a, false, bf, (short)0, acc[s], false, false);
    }
  }
  if (KD4 < KD) {                                    // K-tail (only KD=5)
    const int kb  = KD4 + 2 * half;
    const int k0i = (kb     < KD) ? kb     : KD - 1; // in-bounds addresses
    const int k1i = (kb + 1 < KD) ? kb + 1 : KD - 1;
    v2f bf;
    bf.x = (kb     < KD) ? W[k0i * CO + colc] : 0.f;
    bf.y = (kb + 1 < KD) ? W[k1i * CO + colc] : 0.f;
#pragma unroll
    for (int s = 0; s < 4; ++s) {
      v2f a;
      a.x = (kb     < KD) ? arow[s][k0i] : 0.f;
      a.y = (kb + 1 < KD) ? arow[s][k1i] : 0.f;
      acc[s] = __builtin_amdgcn_wmma_f32_16x16x4_f32(
          false, a, false, bf, (short)0, acc[s], false, false);
    }
  }

  if (col >= CO) return;
  const float bc = bias[colc];
#pragma unroll
  for (int s = 0; s < 4; ++s) {
#pragma unroll
    for (int r = 0; r < 8; ++r) {
      const int m = row0 + 16 * s + 8 * half + r;    // BN%64==0: no row guard
      const int o = m * CO + colc;
      float v = acc[s][r] + bc;
      if (mode == 0) {
        out[o] = v;
      } else {
        v += out[o];
        out[o] = (mode == 1) ? (1.f / (1.f + expf(-v))) : tanhf(v);
      }
    }
  }
}

// ------------------------- pointwise kernels -------------------------
__global__ void k_fill0(float* p, int n) {
  for (int i = blockIdx.x * blockDim.x + threadIdx.x; i < n;
       i += gridDim.x * blockDim.x) p[i] = 0.f;
}
// dst[r*ldd + c] = src[r*lds + c]; cin = 1<<cinL2 (stage-0 identity hop)
__global__ void k_copy_cols(const float* __restrict__ src, int lds_,
                            float* __restrict__ dst, int ldd, int cinL2, int n) {
  const int cm = (1 << cinL2) - 1;
  for (int i = blockIdx.x * blockDim.x + threadIdx.x; i < n;
       i += gridDim.x * blockDim.x) {
    int r = i >> cinL2, c = i & cm;
    dst[r * ldd + c] = src[r * lds_ + c];
  }
}
__global__ void k_mul_inplace(float* __restrict__ a,
                              const float* __restrict__ b, int n) {
  for (int i = blockIdx.x * blockDim.x + threadIdx.x; i < n;
       i += gridDim.x * blockDim.x) a[i] *= b[i];
}
// h = (1-z)*h + z*ht
__global__ void k_gru_update(float* __restrict__ h, const float* __restrict__ z,
                             const float* __restrict__ ht, int n) {
  for (int i = blockIdx.x * blockDim.x + threadIdx.x; i < n;
       i += gridDim.x * blockDim.x) {
    float zi = z[i];
    h[i] = (1.f - zi) * h[i] + zi * ht[i];
  }
}
// dst[b*N+n] = X[(b*T_IN + t)*N + n]
__global__ void k_gather_t(const float* __restrict__ X,
                           float* __restrict__ dst, int t) {
  for (int i = blockIdx.x * blockDim.x + threadIdx.x; i < BN;
       i += gridDim.x * blockDim.x) {
    int b = i / N_NODES, n = i - b * N_NODES;
    dst[i] = X[(b * T_IN + t) * N_NODES + n];
  }
}
// out[(b*T_OUT + t)*N + n] = src[b*N+n]
__global__ void k_scatter_t(float* __restrict__ out,
                            const float* __restrict__ src, int t) {
  for (int i = blockIdx.x * blockDim.x + threadIdx.x; i < BN;
       i += gridDim.x * blockDim.x) {
    int b = i / N_NODES, n = i - b * N_NODES;
    out[(b * T_OUT + t) * N_NODES + n] = src[i];
  }
}

// ----------------------------- host side -----------------------------
struct CellP {
  const float *Whh,*bhh,*Whr,*bhr,*Whz,*bhz,*Wxh,*bxh,*Wxr,*bxr,*Wxz,*bxz;
};
// pytree (sorted-key) leaf order within a cell: hh,hr,hz,xh,xr,xz; W before b
static CellP getCell(void* const* d_in, int base) {
  CellP c;
  c.Whh = (const float*)d_in[base + 0];  c.bhh = (const float*)d_in[base + 1];
  c.Whr = (const float*)d_in[base + 2];  c.bhr = (const float*)d_in[base + 3];
  c.Whz = (const float*)d_in[base + 4];  c.bhz = (const float*)d_in[base + 5];
  c.Wxh = (const float*)d_in[base + 6];  c.bxh = (const float*)d_in[base + 7];
  c.Wxr = (const float*)d_in[base + 8];  c.bxr = (const float*)d_in[base + 9];
  c.Wxz = (const float*)d_in[base +10];  c.bxz = (const float*)d_in[base +11];
  return c;
}

static inline int nblk(int n) {
  int b = (n + 255) / 256; return b < 1 ? 1 : (b > 4096 ? 4096 : b);
}

// Concat diffusion features F (BN x 5*cin): [x, Pf x, Pf^2 x, Pb x, Pb^2 x]
static void diffuse_feats(const float* x, int ldx, int cin, int cinL2, float* F,
                          const float* Pf, const float* Pb, hipStream_t s) {
  const int ldF = 5 * cin;
  k_copy_cols<<<nblk(BN * cin), 256, 0, s>>>(x, ldx, F, ldF, cinL2, BN * cin);
  dim3 blk(32, (cin + 15) / 16);
  dim3 grd((N_NODES + 63) / 64, 1, B_SZ);
  k_diffuse<<<grd, blk, 0, s>>>(Pf, x,           ldx, F + 1 * cin, ldF, cin);
  k_diffuse<<<grd, blk, 0, s>>>(Pf, F + 1 * cin, ldF, F + 2 * cin, ldF, cin);
  k_diffuse<<<grd, blk, 0, s>>>(Pb, x,           ldx, F + 3 * cin, ldF, cin);
  k_diffuse<<<grd, blk, 0, s>>>(Pb, F + 3 * cin, ldF, F + 4 * cin, ldF, cin);
}

static void gemm(const float* A, int KD, const float* W, const float* b, int CO,
                 float* out, int mode, hipStream_t s) {
  dim3 blk(32, (CO + 15) / 16);
  dim3 grd(BN / 64, 1, 1);
  k_gemm_bias_act<<<grd, blk, 0, s>>>(A, KD, W, b, CO, out, mode);
}

// One DCGRU cell: h <- (1-z)*h + z*htilde   (updates h in place)
static void cell_fwd(const float* x, int ldx, int cin, int cinL2, float* h,
                     const CellP& p, float* Fx, float* Fh, float* tz, float* tr,
                     const float* Pf, const float* Pb, hipStream_t s) {
  diffuse_feats(x, ldx, cin, cinL2, Fx, Pf, Pb, s);   // D(x)
  diffuse_feats(h, HIDC, HIDC, 6, Fh, Pf, Pb, s);     // D(h)
  gemm(Fx, 5 * cin, p.Wxz, p.bxz, HIDC, tz, 0, s);    // z pre
  gemm(Fh, 5 * HIDC, p.Whz, p.bhz, HIDC, tz, 1, s);   // z = sigmoid(.)
  gemm(Fx, 5 * cin, p.Wxr, p.bxr, HIDC, tr, 0, s);    // r pre
  gemm(Fh, 5 * HIDC, p.Whr, p.bhr, HIDC, tr, 1, s);   // r = sigmoid(.)
  k_mul_inplace<<<nblk(BN * HIDC), 256, 0, s>>>(tr, h, BN * HIDC); // tr = r*h
  diffuse_feats(tr, HIDC, HIDC, 6, Fh, Pf, Pb, s);    // Fh = D(r*h)
  gemm(Fx, 5 * cin, p.Wxh, p.bxh, HIDC, tr, 0, s);    // ht pre (tr reused)
  gemm(Fh, 5 * HIDC, p.Whh, p.bhh, HIDC, tr, 2, s);   // ht = tanh(.)
  k_gru_update<<<nblk(BN * HIDC), 256, 0, s>>>(h, tz, tr, BN * HIDC);
}

extern "C" void kernel_launch(void* const* d_in, const int* in_sizes, int n_in,
                              void* d_out, int out_size, void* d_ws,
                              size_t ws_size, hipStream_t stream) {
  (void)in_sizes; (void)n_in; (void)out_size; (void)ws_size;
  // d_in order (pytree sorted keys): X, P_fwd, P_bwd,
  //   dec[0](12), dec[1](12), enc[0](12), enc[1](12), proj.W, proj.b, T_out
  const float* X  = (const float*)d_in[0];
  const float* Pf = (const float*)d_in[1];
  const float* Pb = (const float*)d_in[2];
  CellP dec0 = getCell(d_in, 3);
  CellP dec1 = getCell(d_in, 15);
  CellP enc0 = getCell(d_in, 27);
  CellP enc1 = getCell(d_in, 39);
  const float* Wp = (const float*)d_in[51];
  const float* bp = (const float*)d_in[52];
  float* out = (float*)d_out;

  // workspace carve-up (floats)
  float* ws = (float*)d_ws;
  float* Fx = ws;                          // BN*320
  float* Fh = Fx + (size_t)BN * 320;       // BN*320 (reused for D(r*h))
  float* h0 = Fh + (size_t)BN * 320;       // BN*64
  float* h1 = h0 + (size_t)BN * HIDC;      // BN*64
  float* tz = h1 + (size_t)BN * HIDC;      // BN*64
  float* tr = tz + (size_t)BN * HIDC;      // BN*64
  float* x1 = tr + (size_t)BN * HIDC;      // BN (layer-0 input / proj out)

  // ws is poisoned once by the harness: zero recurrent state every call
  k_fill0<<<nblk(2 * BN * HIDC), 256, 0, stream>>>(h0, 2 * BN * HIDC);

  // ---------------- encoder ----------------
  for (int t = 0; t < T_IN; ++t) {
    k_gather_t<<<nblk(BN), 256, 0, stream>>>(X, x1, t);
    cell_fwd(x1, 1, 1, 0,       h0, enc0, Fx, Fh, tz, tr, Pf, Pb, stream);
    cell_fwd(h0, HIDC, HIDC, 6, h1, enc1, Fx, Fh, tz, tr, Pf, Pb, stream);
  }

  // ---------------- decoder ----------------
  k_gather_t<<<nblk(BN), 256, 0, stream>>>(X, x1, T_IN - 1);  // last_input
  for (int t = 0; t < T_OUT; ++t) {
    cell_fwd(x1, 1, 1, 0,       h0, dec0, Fx, Fh, tz, tr, Pf, Pb, stream);
    cell_fwd(h0, HIDC, HIDC, 6, h1, dec1, Fx, Fh, tz, tr, Pf, Pb, stream);
    diffuse_feats(h1, HIDC, HIDC, 6, Fx, Pf, Pb, stream);     // D(dec out)
    gemm(Fx, 5 * HIDC, Wp, bp, 1, x1, 0, stream);             // proj -> x1
    k_scatter_t<<<nblk(BN), 256, 0, stream>>>(out, x1, t);    // out[:, t]
  }
}